// TransitionDown_73839077753155
// MI455X (gfx1250) — compile-verified
//
#include <hip/hip_runtime.h>
#include <math.h>

// Problem constants (match reference)
#define B_   16
#define N_   4096
#define CIN  64
#define COUT 128
#define K_   16
#define M_   1024
#define EPS_ 1e-5f

typedef __attribute__((ext_vector_type(2))) float v2f;
typedef __attribute__((ext_vector_type(4))) float v4f;
typedef __attribute__((ext_vector_type(8))) float v8f;

// ---------------------------------------------------------------------------
// Kernel 0: zero BN accumulators (workspace is poisoned, must re-zero per call)
// ---------------------------------------------------------------------------
__global__ void td_init_kernel(float* gsum, float* gsumsq) {
    int t = threadIdx.x;
    if (t < COUT) { gsum[t] = 0.f; gsumsq[t] = 0.f; }
}

// ---------------------------------------------------------------------------
// Kernel 1: farthest point sampling. One workgroup (256 threads / 8 waves) per
// batch; positions resident in LDS (48KB). Argmax is carried as a packed u64:
//   (float_bits(min_d) << 32) | (N-1-idx)
// distances >= 0 so float bits are unsigned-monotonic; index complement gives
// the reference's lowest-index tie-break. Reduction = plain u64 max.
// ---------------------------------------------------------------------------
__global__ __launch_bounds__(256) void td_fps_kernel(
    const float* __restrict__ pos, int* __restrict__ fps_idx) {
    __shared__ float px[N_], py[N_], pz[N_];
    __shared__ unsigned long long redp[8];
    __shared__ int sel;

    const int b   = blockIdx.x;
    const int tid = threadIdx.x;
    const int lane = tid & 31, wave = tid >> 5;
    const float* p = pos + (size_t)b * N_ * 3;

#pragma unroll
    for (int t = 0; t < 16; ++t) {
        int i = tid + t * 256;
        px[i] = p[i * 3 + 0];
        py[i] = p[i * 3 + 1];
        pz[i] = p[i * 3 + 2];
    }
    float mind[16];
#pragma unroll
    for (int t = 0; t < 16; ++t) mind[t] = 1e10f;

    if (tid == 0) { fps_idx[b * M_ + 0] = 0; sel = 0; }
    __syncthreads();

    int last = 0;
    for (int it = 1; it < M_; ++it) {
        float lx = px[last], ly = py[last], lz = pz[last];
        float bestv = -1e30f; int besti = 0;
#pragma unroll
        for (int t = 0; t < 16; ++t) {
            int i = tid + t * 256;
            float dx = px[i] - lx, dy = py[i] - ly, dz = pz[i] - lz;
            float d = dx * dx + dy * dy + dz * dz;
            mind[t] = fminf(mind[t], d);
            if (mind[t] > bestv) { bestv = mind[t]; besti = i; }
        }
        unsigned long long bp =
            ((unsigned long long)__float_as_uint(bestv) << 32) |
            (unsigned long long)(unsigned)(N_ - 1 - besti);
#pragma unroll
        for (int off = 16; off >= 1; off >>= 1) {
            unsigned long long o = __shfl_xor(bp, off);
            bp = (o > bp) ? o : bp;
        }
        if (lane == 0) redp[wave] = bp;
        __syncthreads();
        if (wave == 0) {
            unsigned long long v = redp[lane & 7];
#pragma unroll
            for (int off = 4; off >= 1; off >>= 1) {
                unsigned long long o = __shfl_xor(v, off);
                v = (o > v) ? o : v;
            }
            if (lane == 0) {
                int ix = N_ - 1 - (int)(unsigned)(v & 0xffffffffull);
                sel = ix;
                fps_idx[b * M_ + it] = ix;
            }
        }
        __syncthreads();
        last = sel;
    }
}

// ---------------------------------------------------------------------------
// Kernel 2: h = features @ W + b via V_WMMA_F32_16X16X4_F32, plus BN partial
// sums. Block = 256 thr (8 waves). Each wave owns a 16-row tile and sweeps all
// 8 column tiles (A-frags cached in registers, W staged in LDS).
// ---------------------------------------------------------------------------
__global__ __launch_bounds__(256) void td_mlp_kernel(
    const float* __restrict__ feat, const float* __restrict__ W,
    const float* __restrict__ bias, float* __restrict__ h,
    float* __restrict__ gsum, float* __restrict__ gsumsq) {
    __shared__ float Wl[CIN * COUT];
    __shared__ float bl[COUT];
    __shared__ float sl[COUT];
    __shared__ float ql[COUT];

    const int tid = threadIdx.x;
    for (int i = tid; i < CIN * COUT; i += 256) Wl[i] = W[i];
    if (tid < COUT) { bl[tid] = bias[tid]; sl[tid] = 0.f; ql[tid] = 0.f; }
    __syncthreads();

    const int lane = tid & 31, wave = tid >> 5;
    const int hi = lane >> 4, col = lane & 15;
    const int row0 = blockIdx.x * 128 + wave * 16;
    const float* arow = feat + (size_t)(row0 + col) * CIN;

    // A fragments: lane covers row (row0+col), K pair (4j+2hi, 4j+2hi+1)
    v2f a[16];
#pragma unroll
    for (int j = 0; j < 16; ++j) a[j] = *(const v2f*)(arow + 4 * j + 2 * hi);

    for (int ct = 0; ct < 8; ++ct) {
        const int c0 = ct * 16;
        v8f acc = {};
#pragma unroll
        for (int j = 0; j < 16; ++j) {
            v2f bf;
            bf.x = Wl[(4 * j + 2 * hi) * COUT + c0 + col];
            bf.y = Wl[(4 * j + 2 * hi + 1) * COUT + c0 + col];
            acc = __builtin_amdgcn_wmma_f32_16x16x4_f32(
                false, a[j], false, bf, (short)0, acc, false, false);
        }
        float bb = bl[c0 + col];
        float s = 0.f, q = 0.f;
#pragma unroll
        for (int i = 0; i < 8; ++i) {
            float v = acc[i] + bb;
            h[(size_t)(row0 + i + 8 * hi) * COUT + c0 + col] = v;
            s += v; q += v * v;
        }
        s += __shfl_xor(s, 16);
        q += __shfl_xor(q, 16);
        if (hi == 0) { atomicAdd(&sl[c0 + col], s); atomicAdd(&ql[c0 + col], q); }
    }
    __syncthreads();
    if (tid < COUT) { atomicAdd(&gsum[tid], sl[tid]); atomicAdd(&gsumsq[tid], ql[tid]); }
}

// ---------------------------------------------------------------------------
// Kernel 3: BN finalize -> per-channel affine (scale, shift)
// ---------------------------------------------------------------------------
__global__ void td_bn_finalize(const float* __restrict__ gsum,
                               const float* __restrict__ gsumsq,
                               const float* __restrict__ gamma,
                               const float* __restrict__ beta,
                               float* __restrict__ scale,
                               float* __restrict__ shift) {
    int c = threadIdx.x;
    const float inv_n = 1.0f / (float)(B_ * N_);
    float mean = gsum[c] * inv_n;
    float var  = gsumsq[c] * inv_n - mean * mean;
    float inv  = rsqrtf(var + EPS_);
    float sc   = inv * gamma[c];
    scale[c] = sc;
    shift[c] = beta[c] - mean * sc;
}

// ---------------------------------------------------------------------------
// Kernel 4: feature-space kNN. One wave per 16-query tile; 256 key tiles.
// q·f via 16 f32 WMMAs per tile; d2 spilled to per-wave LDS tile; two lanes
// per query row maintain sorted top-16 register lists; LDS merge at the end.
// ---------------------------------------------------------------------------
__global__ __launch_bounds__(128) void td_knn_kernel(
    const float* __restrict__ feat, const int* __restrict__ fps_idx,
    int* __restrict__ nbr) {
    __shared__ float qqs[4][16];
    __shared__ float dA [4][256];   // d2 tile, then list-A dists
    __shared__ float dB [4][256];   // list-B dists
    __shared__ int   iA [4][256];
    __shared__ int   iB [4][256];

    const int tid = threadIdx.x, lane = tid & 31, wave = tid >> 5;
    const int hi = lane >> 4, sl = lane & 15;
    const int tile = blockIdx.x * 4 + wave;
    const int b = tile >> 6, mt = tile & 63;
    const float* fb = feat + (size_t)b * N_ * CIN;

    // A fragments (query rows gathered via fps_idx), plus ||q||^2
    const int qrow = fps_idx[b * M_ + mt * 16 + sl];
    const float* qptr = fb + (size_t)qrow * CIN;
    v2f a[16];
    float ssqh = 0.f;
#pragma unroll
    for (int j = 0; j < 16; ++j) {
        v2f t = *(const v2f*)(qptr + 4 * j + 2 * hi);
        a[j] = t;
        ssqh += t.x * t.x + t.y * t.y;
    }
    float qq = ssqh + __shfl_xor(ssqh, 16);
    if (hi == 0) qqs[wave][sl] = qq;
    __syncthreads();
    float qq8[8];
#pragma unroll
    for (int i = 0; i < 8; ++i) qq8[i] = qqs[wave][i + 8 * hi];

    float bestd[16]; int besti[16];
#pragma unroll
    for (int t = 0; t < 16; ++t) { bestd[t] = 3.4e38f; besti[t] = 0; }

    for (int nt = 0; nt < 256; ++nt) {
        const int n0 = nt * 16;
        const float* kptr = fb + (size_t)(n0 + sl) * CIN;
        if (nt < 255) {  // prefetch key row n0+sl+1 (the row after this lane's row)
            __builtin_prefetch(kptr + CIN, 0, 3);
            __builtin_prefetch(kptr + CIN + 32, 0, 3);
        }
        v8f acc = {};
        float ff = 0.f;
#pragma unroll
        for (int c = 0; c < 4; ++c) {
            v4f v0 = *(const v4f*)(kptr + c * 16 + 0);
            v4f v1 = *(const v4f*)(kptr + c * 16 + 4);
            v4f v2 = *(const v4f*)(kptr + c * 16 + 8);
            v4f v3 = *(const v4f*)(kptr + c * 16 + 12);
            ff += v0.x*v0.x + v0.y*v0.y + v0.z*v0.z + v0.w*v0.w;
            ff += v1.x*v1.x + v1.y*v1.y + v1.z*v1.z + v1.w*v1.w;
            ff += v2.x*v2.x + v2.y*v2.y + v2.z*v2.z + v2.w*v2.w;
            ff += v3.x*v3.x + v3.y*v3.y + v3.z*v3.z + v3.w*v3.w;
            v2f b0; b0.x = hi ? v0.z : v0.x; b0.y = hi ? v0.w : v0.y;
            v2f b1; b1.x = hi ? v1.z : v1.x; b1.y = hi ? v1.w : v1.y;
            v2f b2; b2.x = hi ? v2.z : v2.x; b2.y = hi ? v2.w : v2.y;
            v2f b3; b3.x = hi ? v3.z : v3.x; b3.y = hi ? v3.w : v3.y;
            acc = __builtin_amdgcn_wmma_f32_16x16x4_f32(false, a[4*c+0], false, b0, (short)0, acc, false, false);
            acc = __builtin_amdgcn_wmma_f32_16x16x4_f32(false, a[4*c+1], false, b1, (short)0, acc, false, false);
            acc = __builtin_amdgcn_wmma_f32_16x16x4_f32(false, a[4*c+2], false, b2, (short)0, acc, false, false);
            acc = __builtin_amdgcn_wmma_f32_16x16x4_f32(false, a[4*c+3], false, b3, (short)0, acc, false, false);
        }
        // d2 tile -> LDS (C layout: vgpr i => row i+8*hi, col = sl)
#pragma unroll
        for (int i = 0; i < 8; ++i) {
            float d2 = qq8[i] + ff - 2.0f * acc[i];
            dA[wave][(i + 8 * hi) * 16 + sl] = d2;
        }
        // top-K scan: this lane owns query row sl, cols hi*8 .. hi*8+7
#pragma unroll
        for (int j = 0; j < 8; ++j) {
            int cc = hi * 8 + j;
            float d = dA[wave][sl * 16 + cc];
            if (d < bestd[15]) {
                float dv = d; int iv = n0 + cc;
#pragma unroll
                for (int t = 0; t < 16; ++t) {
                    bool cl = dv < bestd[t];
                    float od = bestd[t]; int oi = besti[t];
                    bestd[t] = cl ? dv : od;  besti[t] = cl ? iv : oi;
                    dv = cl ? od : dv;        iv = cl ? oi : iv;
                }
            }
        }
    }

    // merge the two sorted half-lists per query row
#pragma unroll
    for (int t = 0; t < 16; ++t) {
        if (hi == 0) { dA[wave][sl * 16 + t] = bestd[t]; iA[wave][sl * 16 + t] = besti[t]; }
        else         { dB[wave][sl * 16 + t] = bestd[t]; iB[wave][sl * 16 + t] = besti[t]; }
    }
    if (hi == 0) {
        int pa = 0, pb = 0;
        int* outp = nbr + (size_t)(b * M_ + mt * 16 + sl) * K_;
        for (int k = 0; k < K_; ++k) {
            float da = dA[wave][sl * 16 + pa];
            float db = dB[wave][sl * 16 + pb];
            bool ta = (da <= db);
            outp[k] = ta ? iA[wave][sl * 16 + pa] : iB[wave][sl * 16 + pb];
            pa += ta ? 1 : 0;
            pb += ta ? 0 : 1;
        }
    }
}

// ---------------------------------------------------------------------------
// Kernel 5: gather + BN affine + max-pool over K; also positions/batch outs.
// One block per (b,m); 128 threads = COUT channels (coalesced h reads).
// ---------------------------------------------------------------------------
__global__ __launch_bounds__(128) void td_out_kernel(
    const float* __restrict__ h, const int* __restrict__ nbr,
    const int* __restrict__ fps_idx, const float* __restrict__ positions,
    const float* __restrict__ scale, const float* __restrict__ shift,
    float* __restrict__ out) {
    const int bm = blockIdx.x;           // b*M + m
    const int b  = bm >> 10;
    const int c  = threadIdx.x;
    const int* nb = nbr + (size_t)bm * K_;
    float sc = scale[c], sh = shift[c];
    float best = -3.4e38f;
#pragma unroll
    for (int k = 0; k < K_; ++k) {
        int n = nb[k];
        float v = h[(size_t)(b * N_ + n) * COUT + c] * sc + sh;
        best = fmaxf(best, v);
    }
    float* outF = out;
    float* outP = out + (size_t)B_ * M_ * COUT;
    float* outB = out + (size_t)B_ * M_ * COUT + (size_t)B_ * M_ * 3;
    outF[(size_t)bm * COUT + c] = best;
    if (c < 3) {
        int qn = fps_idx[bm];
        outP[(size_t)bm * 3 + c] = positions[(size_t)(b * N_ + qn) * 3 + c];
    }
    if (c == 3) outB[bm] = (float)b;
}

// ---------------------------------------------------------------------------
extern "C" void kernel_launch(void* const* d_in, const int* in_sizes, int n_in,
                              void* d_out, int out_size, void* d_ws, size_t ws_size,
                              hipStream_t stream) {
    (void)in_sizes; (void)n_in; (void)out_size; (void)ws_size;
    const float* features  = (const float*)d_in[0]; // [B,N,CIN]
    const float* positions = (const float*)d_in[1]; // [B,N,3]
    // d_in[2] = batch (int64), values reconstructible from b -- unused
    const float* W     = (const float*)d_in[3];     // [CIN,COUT]
    const float* bias  = (const float*)d_in[4];     // [COUT]
    const float* gamma = (const float*)d_in[5];     // [COUT]
    const float* beta  = (const float*)d_in[6];     // [COUT]
    float* out = (float*)d_out;

    // Workspace carve-up
    char* ws = (char*)d_ws;
    int*   fps_idx = (int*)ws;                                      // B*M ints (64KB)
    float* h       = (float*)(ws + (1 << 16));                      // B*N*COUT f32 (32MB)
    int*   nbr     = (int*)((char*)h + (size_t)B_ * N_ * COUT * 4); // B*M*K ints (1MB)
    float* gsum    = (float*)((char*)nbr + (size_t)B_ * M_ * K_ * 4);
    float* gsumsq  = gsum + COUT;
    float* scale   = gsumsq + COUT;
    float* shift   = scale + COUT;

    td_init_kernel<<<1, 128, 0, stream>>>(gsum, gsumsq);
    td_fps_kernel<<<B_, 256, 0, stream>>>(positions, fps_idx);
    td_mlp_kernel<<<(B_ * N_) / 128, 256, 0, stream>>>(features, W, bias, h, gsum, gsumsq);
    td_bn_finalize<<<1, COUT, 0, stream>>>(gsum, gsumsq, gamma, beta, scale, shift);
    td_knn_kernel<<<(B_ * M_ / 16) / 4, 128, 0, stream>>>(features, fps_idx, nbr);
    td_out_kernel<<<B_ * M_, 128, 0, stream>>>(h, nbr, fps_idx, positions, scale, shift, out);
}